// HausdorffDistanceLoss_82746839924758
// MI455X (gfx1250) — compile-verified
//
#include <hip/hip_runtime.h>
#include <math.h>

// Problem constants (reference: pred [8,1,256,256] f32, target same shape i32)
#define NB   8
#define NH   256
#define NW   256
#define NTOT (NB * NH * NW)     // 524288
#define CAPD 512.0f             // H + W cap from reference _edt
#define EPSF 1e-7f

typedef __attribute__((ext_vector_type(2))) float v2f;
typedef __attribute__((ext_vector_type(8))) float v8f;

__device__ __forceinline__ float sigmoid_f(float x) {
    // numerically stable sigmoid, matches jax.nn.sigmoid semantics closely
    if (x >= 0.0f) {
        return 1.0f / (1.0f + expf(-x));
    } else {
        float e = expf(x);
        return e / (1.0f + e);
    }
}

// ---------------- K0: zero the max accumulators --------------------------
__global__ void hd_init(unsigned* __restrict__ maxv) {
    if (threadIdx.x < 2) maxv[threadIdx.x] = 0u;
}

// ---------------- K1: 1D row pass of the EDT -----------------------------
// One thread per (field, row). Writes capped-and-squared row distances g^2.
__global__ __launch_bounds__(256)
void hd_rowpass(const float* __restrict__ pred, const int* __restrict__ targ,
                float* __restrict__ g2p, float* __restrict__ g2t) {
    int t = blockIdx.x * blockDim.x + threadIdx.x;   // 0 .. 2*NB*NH-1
    if (t >= 2 * NB * NH) return;
    int f = t >> 11;                 // 0: pred-sigmoid field, 1: target field
    int r = t & (NB * NH - 1);       // flattened (b, i) row id
    size_t base = (size_t)r * NW;
    float* out = f ? g2t : g2p;

    // left-to-right: distance to nearest background (==0) on the left
    float last = -1.0e9f;
    for (int j = 0; j < NW; ++j) {
        bool bg;
        if (f) bg = (targ[base + j] == 0);
        else   bg = (sigmoid_f(pred[base + j]) == 0.0f);
        if (bg) last = (float)j;
        out[base + j] = (float)j - last;          // temp: left distance
    }
    // right-to-left: min with right distance, cap, square
    float nxt = 1.0e9f;
    for (int j = NW - 1; j >= 0; --j) {
        bool bg;
        if (f) bg = (targ[base + j] == 0);
        else   bg = (sigmoid_f(pred[base + j]) == 0.0f);
        if (bg) nxt = (float)j;
        float d = fminf(out[base + j], nxt - (float)j);
        d = fminf(d, CAPD);
        out[base + j] = d * d;
    }
}

// ---------------- K2: exact column min-plus pass -------------------------
// dist2[i,j] = min_k (i-k)^2 + g2[k,j].  One block per (field, image, 32-col
// tile); full 256x32 tile staged in LDS (32KB), overwritten in place (safe:
// the tile is fully read into LDS before any global store).
#define TJ 32
__global__ __launch_bounds__(256)
void hd_colpass(float* __restrict__ g2p, float* __restrict__ g2t) {
    __shared__ float4 sg[NH * TJ / 4];          // 2048 float4 = 32 KB

    int blk = blockIdx.x;                        // 0..127
    int f   = blk >> 6;
    int rem = blk & 63;
    int b   = rem >> 3;
    int j0  = (rem & 7) * TJ;
    float* buf  = f ? g2t : g2p;
    size_t base = (size_t)b * NH * NW;

    // cooperative coalesced load of the whole column tile
    #pragma unroll
    for (int it = 0; it < 8; ++it) {
        int l  = threadIdx.x + it * 256;         // 0..2047
        int jj = l & 7;                           // float4 index within row
        int k  = l >> 3;                          // row (k) index
        sg[l] = *(const float4*)(buf + base + (size_t)k * NW + j0 + jj * 4);
    }
    __syncthreads();

    int i = threadIdx.x;                          // output row, 0..255
    float4 acc[8];
    #pragma unroll
    for (int jj = 0; jj < 8; ++jj)
        acc[jj] = make_float4(3.0e38f, 3.0e38f, 3.0e38f, 3.0e38f);

    for (int k = 0; k < NH; ++k) {
        float d  = (float)(i - k);
        float d2 = d * d;                         // hoisted out of j loop
        #pragma unroll
        for (int jj = 0; jj < 8; ++jj) {
            float4 v = sg[k * 8 + jj];            // LDS broadcast (uniform addr)
            acc[jj].x = fminf(acc[jj].x, d2 + v.x);
            acc[jj].y = fminf(acc[jj].y, d2 + v.y);
            acc[jj].z = fminf(acc[jj].z, d2 + v.z);
            acc[jj].w = fminf(acc[jj].w, d2 + v.w);
        }
    }

    // in-place store of squared 2D distances (16B aligned, per-thread float4s)
    #pragma unroll
    for (int jj = 0; jj < 8; ++jj)
        *(float4*)(buf + base + (size_t)i * NW + j0 + jj * 4) = acc[jj];
}

// ---------------- K3: global max of each dist2 field ---------------------
// Non-negative f32 => uint bit pattern is order-preserving => exact,
// order-independent atomicMax (deterministic).
__global__ __launch_bounds__(256)
void hd_max(const float* __restrict__ d2p, const float* __restrict__ d2t,
            unsigned* __restrict__ maxv) {
    int f     = blockIdx.x & 1;
    int slice = blockIdx.x >> 1;                 // 0..127
    const float* buf = f ? d2t : d2p;
    float m = 0.0f;
    #pragma unroll
    for (int it = 0; it < 16; ++it) {
        size_t idx = ((size_t)it * 128 + slice) * 256 + threadIdx.x;
        m = fmaxf(m, buf[idx]);
    }
    __shared__ float red[256];
    int tid = threadIdx.x;
    red[tid] = m;
    __syncthreads();
    for (int s = 128; s > 0; s >>= 1) {
        if (tid < s) red[tid] = fmaxf(red[tid], red[tid + s]);
        __syncthreads();
    }
    if (tid == 0) atomicMax(&maxv[f], __float_as_uint(red[0]));
}

// ---------------- K4: fused loss + WMMA-assisted reduction ---------------
// loss elem = (sigmoid(p)-t)^2 * (d2p/(sqrt(max2p)+eps)^2 + d2t/(...)^2)
// Wave reduction: V_WMMA_F32_16X16X4_F32 with A=ones, B=(ls,0) gives
// D[m][n] = ls[n] + ls[n+16]  (32 partials -> 16, exact f32 adds),
// finished with 4 shfl_xor steps.
__global__ __launch_bounds__(256)
void hd_loss(const float* __restrict__ pred, const int* __restrict__ targ,
             const float* __restrict__ d2p, const float* __restrict__ d2t,
             const unsigned* __restrict__ maxv, float* __restrict__ partials) {
    float sp   = sqrtf(__uint_as_float(maxv[0])) + EPSF;
    float st   = sqrtf(__uint_as_float(maxv[1])) + EPSF;
    float invp = 1.0f / (sp * sp);
    float invt = 1.0f / (st * st);

    int tid  = threadIdx.x;
    size_t g0 = (size_t)blockIdx.x * 256 + tid;
    float ls = 0.0f;
    #pragma unroll
    for (int it = 0; it < 8; ++it) {            // 256 blk * 256 thr * 8 = NTOT
        size_t idx = g0 + (size_t)it * 65536;
        float s = sigmoid_f(pred[idx]);
        float t = (float)targ[idx];
        float e = s - t;
        e = e * e;
        ls += e * (d2p[idx] * invp + d2t[idx] * invt);
    }

    // WMMA 16x16x4 f32: A = ones(16x4), B = per-lane partials in K rows 0/2
    v2f a  = {1.0f, 1.0f};
    v2f bv = {ls, 0.0f};
    v8f c  = {};
    c = __builtin_amdgcn_wmma_f32_16x16x4_f32(
            /*neg_a=*/false, a, /*neg_b=*/false, bv,
            /*c_mod=*/(short)0, c, /*reuse_a=*/false, /*reuse_b=*/false);
    float col = c[0];                            // = ls[n] + ls[n+16]
    #pragma unroll
    for (int m = 1; m <= 8; m <<= 1)             // sum the 16 column sums
        col += __shfl_xor(col, m, 32);

    __shared__ float wsum[8];
    if ((tid & 31) == 0) wsum[tid >> 5] = col;
    __syncthreads();
    if (tid == 0) {
        float s2 = 0.0f;
        #pragma unroll
        for (int w = 0; w < 8; ++w) s2 += wsum[w];
        partials[blockIdx.x] = s2;               // fixed order: deterministic
    }
}

// ---------------- K5: final deterministic reduce + mean ------------------
__global__ __launch_bounds__(256)
void hd_final(const float* __restrict__ partials, float* __restrict__ out) {
    __shared__ float red[256];
    int tid = threadIdx.x;
    red[tid] = partials[tid];
    __syncthreads();
    for (int s = 128; s > 0; s >>= 1) {
        if (tid < s) red[tid] += red[tid + s];
        __syncthreads();
    }
    if (tid == 0) out[0] = red[0] * (1.0f / (float)NTOT);
}

extern "C" void kernel_launch(void* const* d_in, const int* in_sizes, int n_in,
                              void* d_out, int out_size, void* d_ws, size_t ws_size,
                              hipStream_t stream) {
    const float* pred = (const float*)d_in[0];
    const int*   targ = (const int*)d_in[1];

    // workspace layout (floats): [0,N) g2p/d2p (in-place), [N,2N) g2t/d2t,
    // [2N, 2N+2) max bit-patterns, [2N+8, 2N+264) per-block partials
    float*    ws       = (float*)d_ws;
    float*    g2p      = ws;
    float*    g2t      = ws + NTOT;
    unsigned* maxv     = (unsigned*)(ws + 2 * (size_t)NTOT);
    float*    partials = ws + 2 * (size_t)NTOT + 8;

    hd_init   <<<1,   32,  0, stream>>>(maxv);
    hd_rowpass<<<16,  256, 0, stream>>>(pred, targ, g2p, g2t);
    hd_colpass<<<128, 256, 0, stream>>>(g2p, g2t);
    hd_max    <<<256, 256, 0, stream>>>(g2p, g2t, maxv);
    hd_loss   <<<256, 256, 0, stream>>>(pred, targ, g2p, g2t, maxv, partials);
    hd_final  <<<1,   256, 0, stream>>>(partials, (float*)d_out);
}